// Gradient3DHMM_76149770158153
// MI455X (gfx1250) — compile-verified
//
#include <hip/hip_runtime.h>
#include <hip/hip_bf16.h>
#include <math.h>

#define XY 24
#define ZD 8
#define NST (XY * XY * ZD)       // 4608 states
#define NTOK 2000
#define BATCH 8
#define SLEN 10
#define TOKS 16
#define NGRP (SLEN * BATCH)      // 80 (t,b) groups
#define NEGINF (-INFINITY)

typedef __attribute__((ext_vector_type(2))) float v2f;
typedef __attribute__((ext_vector_type(8))) float v8f;

// ---------------------------------------------------------------------------
// 1) prior log-softmax over N states (single block)
// ---------------------------------------------------------------------------
__global__ void k_prior(const float* __restrict__ pw, float* __restrict__ prior_ls) {
    __shared__ float red[1024];
    const int tid = threadIdx.x;
    float m = NEGINF;
    for (int i = tid; i < NST; i += 1024) m = fmaxf(m, pw[i]);
    red[tid] = m; __syncthreads();
    for (int s = 512; s > 0; s >>= 1) {
        if (tid < s) red[tid] = fmaxf(red[tid], red[tid + s]);
        __syncthreads();
    }
    m = red[0]; __syncthreads();
    float sum = 0.f;
    for (int i = tid; i < NST; i += 1024) sum += expf(pw[i] - m);
    red[tid] = sum; __syncthreads();
    for (int s = 512; s > 0; s >>= 1) {
        if (tid < s) red[tid] += red[tid + s];
        __syncthreads();
    }
    const float lse = m + logf(red[0]);
    for (int i = tid; i < NST; i += 1024) prior_ls[i] = pw[i] - lse;
}

// ---------------------------------------------------------------------------
// 2) per-state transition prep: vals[i][j] (log-softmax over first k logits),
//    dest[i][j] (neighbor index or -1), offset order matches the reference.
// ---------------------------------------------------------------------------
__global__ void k_trans(const float* __restrict__ tw,
                        float* __restrict__ vals, int* __restrict__ dst) {
    const int i = blockIdx.x * blockDim.x + threadIdx.x;
    if (i >= NST) return;
    const int z = i / (XY * XY);
    const int r = i - z * (XY * XY);
    const int y = r / XY;
    const int x = r - y * XY;
    const int ox[7] = {0, 1, -1, 0, 0, 0, 0};
    const int oy[7] = {0, 0, 0, 1, -1, 0, 0};
    const int oz[7] = {0, 0, 0, 0, 0, 1, 2};
    bool valid[7];
    int dj[7];
    int k = 0;
    #pragma unroll
    for (int j = 0; j < 7; ++j) {
        const int nx = x + ox[j], ny = y + oy[j], nz = z + oz[j];
        const bool v = (nx >= 0) && (nx < XY) && (ny >= 0) && (ny < XY) && (nz >= 0) && (nz < ZD);
        valid[j] = v;
        dj[j] = nx + XY * (ny + XY * nz);
        k += v ? 1 : 0;
    }
    // logsumexp over tw[i, 0..k-1]
    float m = NEGINF;
    for (int s = 0; s < k; ++s) m = fmaxf(m, tw[i * 7 + s]);
    float sum = 0.f;
    for (int s = 0; s < k; ++s) sum += expf(tw[i * 7 + s] - m);
    const float lsei = m + logf(sum);
    int rr = 0;
    #pragma unroll
    for (int j = 0; j < 7; ++j) {
        if (valid[j]) {
            vals[i * 8 + j] = tw[i * 7 + rr] - lsei;
            dst[i * 8 + j]  = dj[j];
            ++rr;
        } else {
            vals[i * 8 + j] = NEGINF;
            dst[i * 8 + j]  = -1;
        }
    }
}

// ---------------------------------------------------------------------------
// 3) per-row emission logsumexp: lse[i] = logsumexp(ew[i, 0..1999])
// ---------------------------------------------------------------------------
__global__ void k_rowlse(const float* __restrict__ ew, float* __restrict__ lse) {
    __shared__ float red[256];
    const int i = blockIdx.x;
    const int tid = threadIdx.x;
    const float* row = ew + (size_t)i * NTOK;
    float m = NEGINF;
    for (int j = tid; j < NTOK; j += 256) m = fmaxf(m, row[j]);
    red[tid] = m; __syncthreads();
    for (int s = 128; s > 0; s >>= 1) {
        if (tid < s) red[tid] = fmaxf(red[tid], red[tid + s]);
        __syncthreads();
    }
    m = red[0]; __syncthreads();
    float sum = 0.f;
    for (int j = tid; j < NTOK; j += 256) sum += expf(row[j] - m);
    red[tid] = sum; __syncthreads();
    for (int s = 128; s > 0; s >>= 1) {
        if (tid < s) red[tid] += red[tid + s];
        __syncthreads();
    }
    if (tid == 0) lse[i] = m + logf(red[0]);
}

// ---------------------------------------------------------------------------
// 4) build count matrix C[v][g] (2000 x 80) and non-pad counts cnt[g].
//    g = t*8 + b ; tokens = stories[b][t][0..15]. No atomics, deterministic.
// ---------------------------------------------------------------------------
__global__ void k_buildC(const int* __restrict__ stories,
                         float* __restrict__ C, float* __restrict__ cnt) {
    const int idx = blockIdx.x * blockDim.x + threadIdx.x;
    if (idx >= NTOK * NGRP) return;
    const int v = idx / NGRP;
    const int g = idx - v * NGRP;
    const int t = g / BATCH;
    const int b = g - t * BATCH;
    const int* tok = stories + ((size_t)b * SLEN + t) * TOKS;
    int c = 0, nn = 0;
    #pragma unroll
    for (int l = 0; l < TOKS; ++l) {
        const int tv = tok[l];
        c  += (tv == v) ? 1 : 0;
        nn += (tv != NTOK) ? 1 : 0;   // padding token NTOK emits logprob 0
    }
    C[idx] = (float)c;
    if (v == 0) cnt[g] = (float)nn;
}

// ---------------------------------------------------------------------------
// 5) WMMA GEMM: S[g][i] = sum_k ew[i][k]*C[k][g] - cnt[g]*lse[i]
//    One wave32 per 16x16 tile, V_WMMA_F32_16X16X4_F32, K-loop of 500.
//    Layouts per CDNA5 ISA 7.12.2 (32-bit A 16x4 / 32-bit C-D 16x16).
// ---------------------------------------------------------------------------
__global__ void k_gemm(const float* __restrict__ ew, const float* __restrict__ C,
                       const float* __restrict__ lse, const float* __restrict__ cnt,
                       float* __restrict__ S) {
    const int wave = blockIdx.x * (blockDim.x >> 5) + (threadIdx.x >> 5);
    const int lane = threadIdx.x & 31;
    const int MT = NST / 16;             // 288 row tiles
    const int tm = wave % MT;
    const int tg = wave / MT;            // 0..4 (80/16 col tiles)
    if (tg >= NGRP / 16) return;

    const int mrow  = tm * 16 + (lane & 15);          // A: lane -> M
    const int gcol  = tg * 16 + (lane & 15);          // B: lane -> N
    const int khalf = (lane >> 4) * 2;                // lanes 16-31 carry K+2,K+3

    const float* arow = ew + (size_t)mrow * NTOK + khalf;
    const float* bcol = C + (size_t)khalf * NGRP + gcol;

    v8f acc = {0.f, 0.f, 0.f, 0.f, 0.f, 0.f, 0.f, 0.f};
    #pragma unroll 4
    for (int k0 = 0; k0 < NTOK; k0 += 4) {
        v2f a, b;
        a.x = arow[k0];
        a.y = arow[k0 + 1];
        b.x = bcol[(size_t)k0 * NGRP];
        b.y = bcol[(size_t)(k0 + 1) * NGRP];
        acc = __builtin_amdgcn_wmma_f32_16x16x4_f32(
            /*neg_a=*/false, a, /*neg_b=*/false, b,
            /*c_mod=*/(short)0, acc, /*reuse_a=*/false, /*reuse_b=*/false);
    }

    // Epilogue: subtract cnt[g]*lse[i]; D layout: VGPR r -> M = r + 8*(lane>=16)
    const float cg = cnt[gcol];
    const int mbase = tm * 16 + ((lane >> 4) ? 8 : 0);
    #pragma unroll
    for (int r = 0; r < 8; ++r) {
        S[(size_t)gcol * NST + mbase + r] = acc[r] - cg * lse[mbase + r];
    }
}

// ---------------------------------------------------------------------------
// 6) step 0: out[0][b][i] = S[b][i] + prior_ls[i]
// ---------------------------------------------------------------------------
__global__ void k_step0(const float* __restrict__ S, const float* __restrict__ prior_ls,
                        float* __restrict__ out0) {
    const int idx = blockIdx.x * blockDim.x + threadIdx.x;
    if (idx >= BATCH * NST) return;
    const int b = idx / NST;
    const int i = idx - b * NST;
    out0[idx] = S[(size_t)b * NST + i] + prior_ls[i];
}

// ---------------------------------------------------------------------------
// 7) recursion step: out[t][b][i] = S[t*8+b][i] +
//        logsumexp_j( vals[i][j] + out[t-1][b][dest[i][j]] )
// ---------------------------------------------------------------------------
__global__ void k_step(const float* __restrict__ S, const float* __restrict__ vals,
                       const int* __restrict__ dst, const float* __restrict__ outPrev,
                       float* __restrict__ outCur, int g0) {
    const int idx = blockIdx.x * blockDim.x + threadIdx.x;
    if (idx >= BATCH * NST) return;
    const int b = idx / NST;
    const int i = idx - b * NST;
    const float* carry = outPrev + (size_t)b * NST;
    float v[7];
    float m = NEGINF;
    #pragma unroll
    for (int j = 0; j < 7; ++j) {
        const int d = dst[i * 8 + j];
        const float x = (d >= 0) ? (vals[i * 8 + j] + carry[d]) : NEGINF;
        v[j] = x;
        m = fmaxf(m, x);
    }
    float sum = 0.f;
    #pragma unroll
    for (int j = 0; j < 7; ++j) sum += (v[j] > NEGINF) ? expf(v[j] - m) : 0.f;
    outCur[idx] = S[(size_t)(g0 + b) * NST + i] + m + logf(sum);
}

// ---------------------------------------------------------------------------
extern "C" void kernel_launch(void* const* d_in, const int* in_sizes, int n_in,
                              void* d_out, int out_size, void* d_ws, size_t ws_size,
                              hipStream_t stream) {
    (void)in_sizes; (void)n_in; (void)out_size; (void)ws_size;
    const float* tw      = (const float*)d_in[0];   // (N, 7)
    const float* ew      = (const float*)d_in[1];   // (N, 2000)
    const float* pw      = (const float*)d_in[2];   // (N,)
    const int*   stories = (const int*)d_in[3];     // (8, 10, 16)
    float* out = (float*)d_out;                     // (10, 8, N)

    // workspace layout
    float* ws       = (float*)d_ws;
    float* prior_ls = ws;                            // N
    float* vals     = prior_ls + NST;                // N*8
    int*   dst      = (int*)(vals + NST * 8);        // N*8
    float* lse      = (float*)(dst + NST * 8);       // N
    float* Cm       = lse + NST;                     // 2000*80
    float* cnt      = Cm + NTOK * NGRP;              // 80
    float* S        = cnt + NGRP;                    // 80*N

    k_prior<<<1, 1024, 0, stream>>>(pw, prior_ls);
    k_trans<<<(NST + 255) / 256, 256, 0, stream>>>(tw, vals, dst);
    k_rowlse<<<NST, 256, 0, stream>>>(ew, lse);
    k_buildC<<<(NTOK * NGRP + 255) / 256, 256, 0, stream>>>(stories, Cm, cnt);

    // 1440 tiles, 8 waves (256 threads) per block -> 180 blocks exactly
    const int tiles = (NST / 16) * (NGRP / 16);
    k_gemm<<<tiles / 8, 256, 0, stream>>>(ew, Cm, lse, cnt, S);

    const int BN = BATCH * NST;
    k_step0<<<(BN + 255) / 256, 256, 0, stream>>>(S, prior_ls, out);
    for (int t = 1; t < SLEN; ++t) {
        k_step<<<(BN + 255) / 256, 256, 0, stream>>>(
            S, vals, dst,
            out + (size_t)(t - 1) * BN,
            out + (size_t)t * BN,
            t * BATCH);
    }
}